// TopDownHTMM_30356828848686
// MI455X (gfx1250) — compile-verified
//
#include <hip/hip_runtime.h>

// ---------------- CDNA5 wave32 / WMMA helpers ----------------
typedef __attribute__((ext_vector_type(2))) float v2f;
typedef __attribute__((ext_vector_type(8))) float v8f;

#define NN 21845
#define LEAF_START 5461
#define ACC_N 8192

// ws layout (float offsets)
#define O_WN    0L                          // 4*32*32  Wn[p][i][j] = smA[i,j,p]   (B operand, down & eps-num: N=i,K=j)
#define O_WT    4096L                       // 4*32*32  Wt[p][j][i] = smA[i,j,p]   (B operand, up: N=j,K=i)
#define O_WA    8192L                       // 4*32*32  Wa[p][i][j] = smA*log(smA) (B operand, eps-alh: N=i,K=j)
#define O_SMB   12288L                      // 32*256
#define O_SMPI  20480L                      // 32
#define O_LOGPI 20512L                      // 32
#define O_PRIOR 20544L                      // NN*32
#define O_BETA  (O_PRIOR + (long)NN*32)
#define O_TBETA (O_BETA  + (long)NN*32)
#define O_EPS   (O_TBETA + (long)NN*32)
#define O_ACC   (O_EPS   + (long)NN*32)     // ACC_N floats

__device__ __forceinline__ v8f wmma4(v2f a, v2f b, v8f c) {
  // D(16x16,f32) = A(16x4,f32) x B(4x16,f32) + C   -> v_wmma_f32_16x16x4_f32
  return __builtin_amdgcn_wmma_f32_16x16x4_f32(
      /*neg_a=*/false, a, /*neg_b=*/false, b,
      /*c_mod=*/(short)0, c, /*reuse_a=*/false, /*reuse_b=*/false);
}

__device__ __forceinline__ v2f ldv2(const float* p) {
  return *(const v2f*)p;  // 8B-aligned pair load -> global_load_b64
}

__device__ __forceinline__ float wsum32(float v) {
  v += __shfl_xor(v, 16, 32);
  v += __shfl_xor(v, 8, 32);
  v += __shfl_xor(v, 4, 32);
  v += __shfl_xor(v, 2, 32);
  v += __shfl_xor(v, 1, 32);
  return v;
}
__device__ __forceinline__ float hsum16(float v) { // sum within 16-lane half
  v += __shfl_xor(v, 8, 32);
  v += __shfl_xor(v, 4, 32);
  v += __shfl_xor(v, 2, 32);
  v += __shfl_xor(v, 1, 32);
  return v;
}

// ---------------- prep: softmaxes + init ----------------
__global__ void prep_kernel(const float* __restrict__ A, const float* __restrict__ B,
                            const float* __restrict__ Pi, float* __restrict__ ws) {
  float* Wn    = ws + O_WN;
  float* Wt    = ws + O_WT;
  float* Wa    = ws + O_WA;
  float* smB   = ws + O_SMB;
  float* smPi  = ws + O_SMPI;
  float* logPi = ws + O_LOGPI;
  float* prior = ws + O_PRIOR;
  float* acc   = ws + O_ACC;
  int t = threadIdx.x;

  // smA: softmax over i for each (j,p) column; A[i,j,p] at i*128 + j*4 + p
  if (t < 128) {
    int p = t & 3, j = t >> 2;
    float mx = -1e30f;
    for (int i = 0; i < 32; ++i) mx = fmaxf(mx, A[i * 128 + j * 4 + p]);
    float den = 0.f;
    for (int i = 0; i < 32; ++i) den += expf(A[i * 128 + j * 4 + p] - mx);
    float ld = logf(den);
    for (int i = 0; i < 32; ++i) {
      float a = A[i * 128 + j * 4 + p] - mx;
      float s = expf(a) / den;
      Wn[p * 1024 + i * 32 + j] = s;             // natural [i][j]
      Wt[p * 1024 + j * 32 + i] = s;             // transposed [j][i]
      Wa[p * 1024 + i * 32 + j] = s * (a - ld);  // smA * log(smA), natural [i][j]
    }
  }
  // smB: row softmax (rows handled by wave0 lanes, serial over 256 symbols)
  if (t < 32) {
    float mx = -1e30f;
    for (int s2 = 0; s2 < 256; ++s2) mx = fmaxf(mx, B[t * 256 + s2]);
    float den = 0.f;
    for (int s2 = 0; s2 < 256; ++s2) den += expf(B[t * 256 + s2] - mx);
    for (int s2 = 0; s2 < 256; ++s2) smB[t * 256 + s2] = expf(B[t * 256 + s2] - mx) / den;
  }
  // smPi via wave0 shfl reductions; prior[root] = smPi
  if (t < 32) {
    float x = Pi[t];
    float mx = x;
    mx = fmaxf(mx, __shfl_xor(mx, 16, 32));
    mx = fmaxf(mx, __shfl_xor(mx, 8, 32));
    mx = fmaxf(mx, __shfl_xor(mx, 4, 32));
    mx = fmaxf(mx, __shfl_xor(mx, 2, 32));
    mx = fmaxf(mx, __shfl_xor(mx, 1, 32));
    float e = expf(x - mx);
    float den = wsum32(e);
    smPi[t] = e / den;
    logPi[t] = (x - mx) - logf(den);
    prior[t] = e / den;  // prior[node 0]
  }
  for (int k = t; k < ACC_N; k += 256) acc[k] = 0.f;
}

// ---------------- downward prior pass (WMMA) ----------------
// prior[ch][i] = sum_j smA[i,j,pos] * prior[pa][j]
// B-frag element (K=j, N=i) = smA[i,j,p] = Wn[p][i*32 + j]  -> contiguous K pair
__global__ void down_level(const float* __restrict__ Wn, float* __restrict__ prior,
                           int edge_lo, int perPos) {
  int gp = (perPos + 15) >> 4;
  int wave = blockIdx.x * (blockDim.x >> 5) + (threadIdx.x >> 5);
  if (wave >= 4 * gp) return;
  int p = wave / gp, g = wave % gp;
  int lane = threadIdx.x & 31;
  int hf = lane >> 4, ln = lane & 15;

  int t = g * 16 + ln;
  int tc = (t < perPos) ? t : (perPos - 1);
  int edge = edge_lo + p + 4 * tc;
  const float* prow = prior + (edge >> 2) * 32;  // parent prior row
  const float* W0 = Wn + p * 1024 + ln * 32;         // N = ln       tile
  const float* W1 = Wn + p * 1024 + (16 + ln) * 32;  // N = 16 + ln  tile

  v8f c0 = {}, c1 = {};
#pragma unroll
  for (int kb = 0; kb < 8; ++kb) {
    int k0 = kb * 4 + 2 * hf;
    v2f a  = ldv2(prow + k0);
    v2f b0 = ldv2(W0 + k0);
    v2f b1 = ldv2(W1 + k0);
    c0 = wmma4(a, b0, c0);
    c1 = wmma4(a, b1, c1);
  }
#pragma unroll
  for (int r = 0; r < 8; ++r) {
    int m = r + 8 * hf;
    int tt = g * 16 + m;
    if (tt < perPos) {
      int e2 = edge_lo + p + 4 * tt;
      int ch = e2 + 1;
      prior[ch * 32 + ln]      = c0[r];
      prior[ch * 32 + 16 + ln] = c1[r];
    }
  }
}

// ---------------- emission + leaf normalize ----------------
__global__ void emission_kernel(const float* __restrict__ ws_smB, const float* __restrict__ prior,
                                const int* __restrict__ labels, float* __restrict__ beta) {
  int n = blockIdx.x * 8 + (threadIdx.x >> 5);
  if (n >= NN) return;
  int i = threadIdx.x & 31;
  float e = prior[n * 32 + i] * ws_smB[i * 256 + labels[n]];
  if (n >= LEAF_START) {
    float s = wsum32(e);
    e /= s;
  }
  beta[n * 32 + i] = e;
}

// ---------------- upward t_beta pass (WMMA) ----------------
// t_beta[ch][j] = (sum_i beta[ch][i]*smA[i,j,pos]) / prior[pa][j]
// B-frag element (K=i, N=j) = smA[i,j,p] = Wt[p][j*32 + i]  -> contiguous K pair
__global__ void up_level(const float* __restrict__ Wt, const float* __restrict__ beta,
                         const float* __restrict__ prior, float* __restrict__ t_beta,
                         int edge_lo, int perPos) {
  int gp = (perPos + 15) >> 4;
  int wave = blockIdx.x * (blockDim.x >> 5) + (threadIdx.x >> 5);
  if (wave >= 4 * gp) return;
  int p = wave / gp, g = wave % gp;
  int lane = threadIdx.x & 31;
  int hf = lane >> 4, ln = lane & 15;

  int t = g * 16 + ln;
  int tc = (t < perPos) ? t : (perPos - 1);
  int edge = edge_lo + p + 4 * tc;
  const float* brow = beta + (edge + 1) * 32;
  const float* W0 = Wt + p * 1024 + ln * 32;
  const float* W1 = Wt + p * 1024 + (16 + ln) * 32;

  v8f c0 = {}, c1 = {};
#pragma unroll
  for (int kb = 0; kb < 8; ++kb) {
    int k0 = kb * 4 + 2 * hf;
    v2f a  = ldv2(brow + k0);
    v2f b0 = ldv2(W0 + k0);
    v2f b1 = ldv2(W1 + k0);
    c0 = wmma4(a, b0, c0);
    c1 = wmma4(a, b1, c1);
  }
#pragma unroll
  for (int r = 0; r < 8; ++r) {
    int m = r + 8 * hf;
    int tt = g * 16 + m;
    if (tt < perPos) {
      int e2 = edge_lo + p + 4 * tt;
      int ch = e2 + 1, pa = e2 >> 2;
      t_beta[ch * 32 + ln]      = c0[r] / prior[pa * 32 + ln];
      t_beta[ch * 32 + 16 + ln] = c1[r] / prior[pa * 32 + 16 + ln];
    }
  }
}

// ---------------- parent update: beta[pa] *= prod children t_beta; normalize ----------------
__global__ void parent_update(float* __restrict__ beta, const float* __restrict__ t_beta,
                              int pa_lo, int nPa) {
  int w = blockIdx.x * 8 + (threadIdx.x >> 5);
  if (w >= nPa) return;
  int q = pa_lo + w;
  int j = threadIdx.x & 31;
  float v = beta[q * 32 + j];
  v *= t_beta[(4 * q + 1) * 32 + j];
  v *= t_beta[(4 * q + 2) * 32 + j];
  v *= t_beta[(4 * q + 3) * 32 + j];
  v *= t_beta[(4 * q + 4) * 32 + j];
  float s = wsum32(v);
  beta[q * 32 + j] = v / s;
}

__global__ void set_root_eps(float* __restrict__ eps, const float* __restrict__ beta) {
  eps[threadIdx.x] = beta[threadIdx.x];
}

// ---------------- eps downward pass (WMMA x2 matrices) + A_lh partials ----------------
// v[j] = eps[pa][j]/t_beta[ch][j]; f[i] = beta[ch][i]/prior[ch][i]
// num_eps[i] = f[i]*sum_j smA[i,j]*v[j];  alh = sum_i f[i]*sum_j (smA*logsmA)[i,j]*v[j]
__global__ void eps_level(const float* __restrict__ Wn, const float* __restrict__ Wa,
                          const float* __restrict__ beta, const float* __restrict__ prior,
                          const float* __restrict__ t_beta, float* __restrict__ eps,
                          float* __restrict__ acc, int edge_lo, int perPos, int slot_base) {
  int gp = (perPos + 15) >> 4;
  int wave = blockIdx.x * (blockDim.x >> 5) + (threadIdx.x >> 5);
  if (wave >= 4 * gp) return;
  int p = wave / gp, g = wave % gp;
  int lane = threadIdx.x & 31;
  int hf = lane >> 4, ln = lane & 15;

  int t = g * 16 + ln;
  int tc = (t < perPos) ? t : (perPos - 1);
  int edge = edge_lo + p + 4 * tc;
  const float* er = eps + (edge >> 2) * 32;
  const float* tb = t_beta + (edge + 1) * 32;
  const float* N0 = Wn + p * 1024 + ln * 32;
  const float* N1 = Wn + p * 1024 + (16 + ln) * 32;
  const float* A0 = Wa + p * 1024 + ln * 32;
  const float* A1 = Wa + p * 1024 + (16 + ln) * 32;

  v8f ce0 = {}, ce1 = {}, ca0 = {}, ca1 = {};
#pragma unroll
  for (int kb = 0; kb < 8; ++kb) {
    int k0 = kb * 4 + 2 * hf;
    v2f ev = ldv2(er + k0);
    v2f tv = ldv2(tb + k0);
    v2f a = ev / tv;
    v2f b0 = ldv2(N0 + k0);
    v2f b1 = ldv2(N1 + k0);
    v2f d0 = ldv2(A0 + k0);
    v2f d1 = ldv2(A1 + k0);
    ce0 = wmma4(a, b0, ce0);
    ce1 = wmma4(a, b1, ce1);
    ca0 = wmma4(a, d0, ca0);
    ca1 = wmma4(a, d1, ca1);
  }
  float alh = 0.f;
#pragma unroll
  for (int r = 0; r < 8; ++r) {
    int m = r + 8 * hf;
    int tt = g * 16 + m;
    int ttc = (tt < perPos) ? tt : (perPos - 1);
    int e2 = edge_lo + p + 4 * ttc;
    int c2 = e2 + 1;
    float f0 = beta[c2 * 32 + ln]      / prior[c2 * 32 + ln];
    float f1 = beta[c2 * 32 + 16 + ln] / prior[c2 * 32 + 16 + ln];
    float de0 = ce0[r] * f0, de1 = ce1[r] * f1;
    float s = hsum16(de0 + de1);  // row sum over all 32 states (both N-tiles, 16 lanes)
    if (tt < perPos) {
      eps[c2 * 32 + ln]      = de0 / s;
      eps[c2 * 32 + 16 + ln] = de1 / s;
      alh += ca0[r] * f0 + ca1[r] * f1;
    }
  }
  alh = wsum32(alh);
  if (lane == 0) acc[slot_base + p * gp + g] = alh;
}

// ---------------- B_lh partials ----------------
__global__ void blh_kernel(const float* __restrict__ ws_smB, const float* __restrict__ eps,
                           const int* __restrict__ labels, float* __restrict__ acc, int base) {
  __shared__ float sred[8];
  int w = threadIdx.x >> 5;
  int n = blockIdx.x * 8 + w;
  int i = threadIdx.x & 31;
  float v = 0.f;
  if (n < NN) v = eps[n * 32 + i] * logf(ws_smB[i * 256 + labels[n]]);
  v = wsum32(v);
  if (i == 0) sred[w] = v;
  __syncthreads();
  if (threadIdx.x == 0) {
    float s = 0.f;
    for (int k = 0; k < 8; ++k) s += sred[k];
    acc[base + blockIdx.x] = s;
  }
}

// ---------------- final fixed-order reduction ----------------
__global__ void final_reduce(const float* __restrict__ acc, int n, const float* __restrict__ eps,
                             const float* __restrict__ logPi, float* __restrict__ out) {
  __shared__ float sred[256];
  int t = threadIdx.x;
  float s = 0.f;
  for (int k = t; k < n; k += 256) s += acc[k];
  if (t < 32) s += eps[t] * logPi[t];  // Pi_lh from root eps
  sred[t] = s;
  __syncthreads();
  for (int off = 128; off >= 1; off >>= 1) {
    if (t < off) sred[t] += sred[t + off];
    __syncthreads();
  }
  if (t == 0) out[0] = sred[0];
}

// ---------------- host launcher ----------------
extern "C" void kernel_launch(void* const* d_in, const int* in_sizes, int n_in,
                              void* d_out, int out_size, void* d_ws, size_t ws_size,
                              hipStream_t stream) {
  (void)in_sizes; (void)n_in; (void)out_size; (void)ws_size;
  static const int OFFS[8] = {0, 4, 20, 84, 340, 1364, 5460, 21844};

  const float* A      = (const float*)d_in[0];
  const float* B      = (const float*)d_in[1];
  const float* Pi     = (const float*)d_in[2];
  const int*   labels = (const int*)d_in[5];
  float* out = (float*)d_out;
  float* ws  = (float*)d_ws;

  float* Wn    = ws + O_WN;
  float* Wt    = ws + O_WT;
  float* Wa    = ws + O_WA;
  float* smB   = ws + O_SMB;
  float* logPi = ws + O_LOGPI;
  float* prior = ws + O_PRIOR;
  float* beta  = ws + O_BETA;
  float* tbeta = ws + O_TBETA;
  float* eps   = ws + O_EPS;
  float* acc   = ws + O_ACC;

  prep_kernel<<<1, 256, 0, stream>>>(A, B, Pi, ws);

  // downward prior pass
  for (int lv = 0; lv < 7; ++lv) {
    int lo = OFFS[lv], pp = (OFFS[lv + 1] - OFFS[lv]) / 4;
    int gp = (pp + 15) / 16, waves = 4 * gp, blocks = (waves + 7) / 8;
    down_level<<<blocks, 256, 0, stream>>>(Wn, prior, lo, pp);
  }

  // beta = prior * emission; normalize leaves
  emission_kernel<<<(NN + 7) / 8, 256, 0, stream>>>(smB, prior, labels, beta);

  // upward pass
  for (int lv = 6; lv >= 0; --lv) {
    int lo = OFFS[lv], pp = (OFFS[lv + 1] - OFFS[lv]) / 4;
    int gp = (pp + 15) / 16, waves = 4 * gp, blocks = (waves + 7) / 8;
    up_level<<<blocks, 256, 0, stream>>>(Wt, beta, prior, tbeta, lo, pp);
    int nPa = pp, paLo = lo >> 2;
    parent_update<<<(nPa + 7) / 8, 256, 0, stream>>>(beta, tbeta, paLo, nPa);
  }

  set_root_eps<<<1, 32, 0, stream>>>(eps, beta);

  // eps downward pass + A_lh partials
  int slot = 0;
  for (int lv = 0; lv < 7; ++lv) {
    int lo = OFFS[lv], pp = (OFFS[lv + 1] - OFFS[lv]) / 4;
    int gp = (pp + 15) / 16, waves = 4 * gp, blocks = (waves + 7) / 8;
    eps_level<<<blocks, 256, 0, stream>>>(Wn, Wa, beta, prior, tbeta, eps, acc, lo, pp, slot);
    slot += 4 * gp;
  }

  // B_lh partials
  int blhBase = slot;
  int blocksB = (NN + 7) / 8;
  blh_kernel<<<blocksB, 256, 0, stream>>>(smB, eps, labels, acc, blhBase);

  final_reduce<<<1, 256, 0, stream>>>(acc, blhBase + blocksB, eps, logPi, out);
}